// YOLOLoss_48962627174832
// MI455X (gfx1250) — compile-verified
//
#include <hip/hip_runtime.h>

// ---------------- problem constants (match reference) ----------------
namespace {
constexpr int S_GRID = 7;
constexpr int NBOX   = 2;            // B
constexpr int NCLS   = 20;           // C
constexpr int REC    = NBOX * (5 + NCLS);   // 50 floats per cell
constexpr int NSAMP  = 8192;
constexpr int NCELLS = NSAMP * S_GRID * S_GRID;   // 401408
constexpr int CPB    = 128;                       // cells per block (4 waves)
constexpr int NBLK   = NCELLS / CPB;              // 3136 (exact)
constexpr int BYTES_PER_ARR = CPB * REC * 4;      // 25600 B per array per block
constexpr int XFERS  = 2 * BYTES_PER_ARR / 16;    // 3200 b128 transfers per block
constexpr int HALF_X = XFERS / 2;                 // 1600 (preds half)
constexpr float LAMBDA_NOOBJ = 0.5f;
constexpr float EPS_F = 1e-9f;
}

__device__ __forceinline__ float iou_xywh(const float* a, const float* b) {
    float ax1 = a[0] - a[2] * 0.5f, ay1 = a[1] - a[3] * 0.5f;
    float ax2 = a[0] + a[2] * 0.5f, ay2 = a[1] + a[3] * 0.5f;
    float bx1 = b[0] - b[2] * 0.5f, by1 = b[1] - b[3] * 0.5f;
    float bx2 = b[0] + b[2] * 0.5f, by2 = b[1] + b[3] * 0.5f;
    float iw = fmaxf(fminf(ax2, bx2) - fmaxf(ax1, bx1), 0.0f);
    float ih = fmaxf(fminf(ay2, by2) - fmaxf(ay1, by1), 0.0f);
    float inter = iw * ih;
    float uni = a[2] * a[3] + b[2] * b[3] - inter;
    return inter / (uni + EPS_F);
}

// ------------- pass 1: per-block partial loss via async-LDS staging -------------
__global__ __launch_bounds__(CPB) void yolo_partial_kernel(
    const float* __restrict__ preds,
    const float* __restrict__ targets,
    float* __restrict__ partials) {
    __shared__ __align__(16) float lds[2 * CPB * REC];   // 51200 B: preds | targets
    __shared__ float red[CPB];

    const int tid = threadIdx.x;
    const int blk = blockIdx.x;

    // LDS byte offset of our staging buffer (low 32 bits of the flat address).
    unsigned lds_base = (unsigned)(unsigned long long)&lds[0];

    unsigned long long pbase = (unsigned long long)preds   + (unsigned long long)blk * BYTES_PER_ARR;
    unsigned long long tbase = (unsigned long long)targets + (unsigned long long)blk * BYTES_PER_ARR;
    // second half of linear transfer space maps to targets:
    unsigned long long tbase_shift = tbase - (unsigned long long)HALF_X * 16ull;

    // 25 fully-coalesced async b128 transfers per thread (CDNA5 async path, ASYNCcnt).
#pragma unroll
    for (int it = 0; it < XFERS / CPB; ++it) {
        int linear = it * CPB + tid;                       // 0 .. 3199
        unsigned lds_dst = lds_base + (unsigned)linear * 16u;
        unsigned long long g =
            (linear < HALF_X ? pbase : tbase_shift) + (unsigned long long)linear * 16ull;
        asm volatile("global_load_async_to_lds_b128 %0, %1, off"
                     :
                     : "v"(lds_dst), "v"(g)
                     : "memory");
    }
    asm volatile("s_wait_asynccnt 0x0" ::: "memory");
    __syncthreads();

    // Each thread processes one cell straight out of LDS.
    const float* p  = &lds[tid * REC];
    const float* t  = &lds[CPB * REC + tid * REC];
    const float* p0 = p;        const float* p1 = p + 25;
    const float* t0 = t;        const float* t1 = t + 25;

    float iou0 = iou_xywh(p0, t0);
    float iou1 = iou_xywh(p1, t1);
    int   best = (iou1 > iou0) ? 1 : 0;                 // argmax, first index on tie
    bool  has_obj = (t0[4] > 0.0f) || (t1[4] > 0.0f);

    float d0x = p0[0] - t0[0], d0y = p0[1] - t0[1];
    float d1x = p1[0] - t1[0], d1y = p1[1] - t1[1];
    float dxy0 = d0x * d0x + d0y * d0y;
    float dxy1 = d1x * d1x + d1y * d1y;
    float loss = has_obj ? (best ? dxy1 : dxy0) : 0.0f;

    // class loss + gt-class argmax over the LAST box's class scores
    const float* pc = p1 + 5;
    const float* tc = t1 + 5;
    float mx = tc[0];
    int   gi = 0;
    float cls = 0.0f;
#pragma unroll
    for (int c = 0; c < NCLS; ++c) {
        float d = pc[c] - tc[c];
        cls += d * d;
        if (c > 0 && tc[c] > mx) { mx = tc[c]; gi = c; }
    }
    loss += cls;

    float pg  = pc[gi];
    float ce0 = iou0 * pg - iou0; ce0 *= ce0;
    float ce1 = iou1 * pg - iou1; ce1 *= ce1;
    float w0  = (best == 0) ? 1.0f : LAMBDA_NOOBJ;
    float w1  = (best == 1) ? 1.0f : LAMBDA_NOOBJ;
    loss += w0 * ce0 + w1 * ce1;

    // deterministic block tree reduction
    red[tid] = loss;
    __syncthreads();
#pragma unroll
    for (int s2 = CPB / 2; s2 > 0; s2 >>= 1) {
        if (tid < s2) red[tid] += red[tid + s2];
        __syncthreads();
    }
    if (tid == 0) partials[blk] = red[0];
}

// ------------- pass 2: single-wave deterministic final reduction (WMMA) -------------
typedef float v2f __attribute__((ext_vector_type(2)));
typedef float v8f __attribute__((ext_vector_type(8)));

__global__ __launch_bounds__(32) void yolo_reduce_kernel(
    const float* __restrict__ partials, int n, float* __restrict__ out, float inv_n) {
    const int lane = threadIdx.x;   // exactly one wave32, EXEC all ones
    float acc = 0.0f;
    for (int i = lane; i < n; i += 32) acc += partials[i];   // n % 32 == 0: uniform trips

    float tot;
#if __has_builtin(__builtin_amdgcn_wmma_f32_16x16x4_f32)
    // A (16x4 f32): lane L<16 -> {M=L,K=0/1}, lane L>=16 -> {M=L-16,K=2/3}.
    // Set A.v0 = acc, A.v1 = 0; B = all ones  =>  D[m][n] = acc[m] + acc[m+16].
    v2f a; a.x = acc;  a.y = 0.0f;
    v2f b; b.x = 1.0f; b.y = 1.0f;
    v8f c = {};
    v8f d = __builtin_amdgcn_wmma_f32_16x16x4_f32(
        /*neg_a=*/false, a, /*neg_b=*/false, b,
        /*c_mod=*/(short)0, c, /*reuse_a=*/false, /*reuse_b=*/false);
    // lane 0 holds D[0..7][0] in d[0..7]; lane 16 holds D[8..15][0].
    float s = d[0] + d[1] + d[2] + d[3] + d[4] + d[5] + d[6] + d[7];
    tot = __shfl(s, 0, 32) + __shfl(s, 16, 32);
#else
    // fallback: butterfly shuffle reduction
    for (int off = 16; off > 0; off >>= 1) acc += __shfl_xor(acc, off, 32);
    tot = acc;
#endif
    if (lane == 0) out[0] = tot * inv_n;
}

// ----------------------------- launch -----------------------------
extern "C" void kernel_launch(void* const* d_in, const int* in_sizes, int n_in,
                              void* d_out, int out_size, void* d_ws, size_t ws_size,
                              hipStream_t stream) {
    const float* preds   = (const float*)d_in[0];
    const float* targets = (const float*)d_in[1];
    float* out      = (float*)d_out;
    float* partials = (float*)d_ws;   // NBLK floats (12.5 KB) of scratch

    yolo_partial_kernel<<<NBLK, CPB, 0, stream>>>(preds, targets, partials);
    yolo_reduce_kernel<<<1, 32, 0, stream>>>(partials, NBLK, out, 1.0f / (float)NSAMP);
}